// OrthogonalLayerHouseholder1D_59313498357836
// MI455X (gfx1250) — compile-verified
//
#include <hip/hip_runtime.h>

typedef float v2f __attribute__((ext_vector_type(2)));
typedef float v8f __attribute__((ext_vector_type(8)));

#define N_MODELS  4
#define N_SAMPLES 8192
#define DIM       1024
#define STR       1040   // padded LDS row stride (floats): shifts banks by 16/model

__global__ __launch_bounds__(256) void cholqr_wmma_kernel(const float* __restrict__ x,
                                                          float* __restrict__ out) {
    __shared__ float vbuf[N_MODELS * STR];   // staged sample tile, 16.6 KB
    __shared__ float gram[10];               // upper-tri Gram accumulator

    const int s    = blockIdx.x;
    const int t    = threadIdx.x;
    const int lane = t & 31;
    const int wave = t >> 5;

    if (t < 10) gram[t] = 0.0f;

    const size_t SD = (size_t)N_SAMPLES * DIM;
    const size_t rowbase = (size_t)s * DIM;
    const int d0 = t * 4;

    // ---- Phase 1: coalesced b128 loads, partial Gram, stage to LDS ----
    float4 v[4];
#pragma unroll
    for (int m = 0; m < 4; ++m) {
        v[m] = *(const float4*)(x + (size_t)m * SD + rowbase + d0);
        *(float4*)(&vbuf[m * STR + d0]) = v[m];
    }

    const int pa[10] = {0,0,0,0,1,1,1,2,2,3};
    const int pb[10] = {0,1,2,3,1,2,3,2,3,3};
    float p[10];
#pragma unroll
    for (int k = 0; k < 10; ++k) {
        float4 a = v[pa[k]], b = v[pb[k]];
        p[k] = a.x*b.x + a.y*b.y + a.z*b.z + a.w*b.w;
    }
    // wave32 butterfly reduction
#pragma unroll
    for (int off = 16; off > 0; off >>= 1) {
#pragma unroll
        for (int k = 0; k < 10; ++k) p[k] += __shfl_xor(p[k], off, 32);
    }
    __syncthreads();                      // gram zeroed + vbuf fully written
    if (lane == 0) {
#pragma unroll
        for (int k = 0; k < 10; ++k) atomicAdd(&gram[k], p[k]);  // ds_add_f32
    }
    __syncthreads();                      // Gram complete

    // ---- Phase 2: redundant per-thread 4x4 Cholesky + triangular inverse ----
    float g[10];
#pragma unroll
    for (int k = 0; k < 10; ++k) g[k] = gram[k];
    float G[4][4];
    G[0][0]=g[0]; G[0][1]=g[1]; G[0][2]=g[2]; G[0][3]=g[3];
    G[1][0]=g[1]; G[1][1]=g[4]; G[1][2]=g[5]; G[1][3]=g[6];
    G[2][0]=g[2]; G[2][1]=g[5]; G[2][2]=g[7]; G[2][3]=g[8];
    G[3][0]=g[3]; G[3][1]=g[6]; G[3][2]=g[8]; G[3][3]=g[9];

    float L[4][4];
#pragma unroll
    for (int i = 0; i < 4; ++i)
#pragma unroll
        for (int j = 0; j < 4; ++j) L[i][j] = 0.0f;
#pragma unroll
    for (int j = 0; j < 4; ++j) {
        float ss = G[j][j];
#pragma unroll
        for (int k = 0; k < j; ++k) ss -= L[j][k] * L[j][k];
        float dj = sqrtf(fmaxf(ss, 1.175494e-38f));
        L[j][j] = dj;
        float rdj = 1.0f / dj;
#pragma unroll
        for (int i = j + 1; i < 4; ++i) {
            float tt = G[i][j];
#pragma unroll
            for (int k = 0; k < j; ++k) tt -= L[i][k] * L[j][k];
            L[i][j] = tt * rdj;
        }
    }
    float Li[4][4];                       // Li = inv(L), lower triangular
#pragma unroll
    for (int i = 0; i < 4; ++i)
#pragma unroll
        for (int j = 0; j < 4; ++j) Li[i][j] = 0.0f;
#pragma unroll
    for (int j = 0; j < 4; ++j) {
        Li[j][j] = 1.0f / L[j][j];
#pragma unroll
        for (int i = j + 1; i < 4; ++i) {
            float acc = 0.0f;
#pragma unroll
            for (int k = j; k < i; ++k) acc += L[i][k] * Li[k][j];
            Li[i][j] = -acc / L[i][i];
        }
    }
    // W = L^{-T} (upper triangular): Q = A * W.  A-frag needs Afrag[M][K] = W[K][M].
    float W[4][4];
#pragma unroll
    for (int i = 0; i < 4; ++i)
#pragma unroll
        for (int j = 0; j < 4; ++j) W[i][j] = Li[j][i];

    // ---- Phase 3: Q^T chunk = W^T(4x4 in 16x4) x V^T(4x16) via V_WMMA_F32_16X16X4_F32 ----
    // Fragment layouts per ISA 7.12.2 (f32, wave32):
    //   A 16x4: lanes 0-15 M=0..15 | lanes 16-31 M=0..15; VGPR0: K=0|K=2, VGPR1: K=1|K=3
    //   B 4x16: VGPR0: row K=0 (lanes 0-15) | row K=2 (lanes 16-31); VGPR1: K=1 | K=3
    //   D 16x16: VGPR r: row M=r (lanes 0-15) | row M=8+r (lanes 16-31)
    const int col = lane & 15;
    const int hi  = lane >> 4;
    const int k0  = hi ? 2 : 0;
    const int k1  = hi ? 3 : 1;

    v2f afrag;
    afrag.x = (col < 4) ? W[k0][col] : 0.0f;
    afrag.y = (col < 4) ? W[k1][col] : 0.0f;

    for (int c = wave; c < DIM / 16; c += 8) {   // uniform per wave -> EXEC all-ones at WMMA
        const int base = c * 16;
        v2f bfrag;
        bfrag.x = vbuf[k0 * STR + base + col];   // bank-conflict-free (padded stride)
        bfrag.y = vbuf[k1 * STR + base + col];

        v8f acc = {};
        acc = __builtin_amdgcn_wmma_f32_16x16x4_f32(
            /*neg_a=*/false, afrag, /*neg_b=*/false, bfrag,
            /*c_mod=*/(short)0, acc, /*reuse_a=*/false, /*reuse_b=*/false);

        // Pair rows {0,2} and {1,3} into full-wave stores: upper half fetches rows 2/3
        // from lanes 0-15 via bpermute.
        float t2 = __shfl(acc[2], col, 32);
        float t3 = __shfl(acc[3], col, 32);
        float val0 = hi ? t2 : acc[0];
        float val1 = hi ? t3 : acc[1];
        const size_t dpos = rowbase + (size_t)(base + col);
        out[(size_t)k0 * SD + dpos] = val0;      // models 0|2
        out[(size_t)k1 * SD + dpos] = val1;      // models 1|3
    }
}

extern "C" void kernel_launch(void* const* d_in, const int* in_sizes, int n_in,
                              void* d_out, int out_size, void* d_ws, size_t ws_size,
                              hipStream_t stream) {
    (void)in_sizes; (void)n_in; (void)out_size; (void)d_ws; (void)ws_size;
    const float* x = (const float*)d_in[0];
    float* out = (float*)d_out;
    hipLaunchKernelGGL(cholqr_wmma_kernel, dim3(N_SAMPLES), dim3(256), 0, stream, x, out);
}